// TvarLayer_70927089926375
// MI455X (gfx1250) — compile-verified
//
#include <hip/hip_runtime.h>

typedef __attribute__((ext_vector_type(2))) float v2f;
typedef __attribute__((ext_vector_type(8))) float v8f;

#define NN 16
#define CCH 16
#define OO 16
#define HH 52
#define WWD 52
#define KP 9           // k*k patch size
#define OH 50
#define MTOT 22500.0f  // KP * OH*OH

// ---------------------------------------------------------------------------
// Stage 1: per (n,c) image, compute the 9 windowed sums of x (A1) and x^2 (A2).
// Windowed sum over 50x50 region at offset (i,j) of the 52x52 image.
// Row trick: rowWin[r][j] = rowTotal[r] - edge elements; then 9 column sums.
// ---------------------------------------------------------------------------
__global__ void tvar_stage1(const float* __restrict__ x,
                            float* __restrict__ a1,
                            float* __restrict__ a2) {
  __shared__ float rw1[3][HH];
  __shared__ float rw2[3][HH];
  const int nc = blockIdx.x;                       // n*16 + c
  const float* img = x + (size_t)nc * (HH * WWD);
  const int t = threadIdx.x;                       // 64 threads

  if (t < HH) {
    const float* row = img + t * WWD;
    float s1 = 0.f, s2 = 0.f;
    const float e0 = row[0], e1 = row[1], e50 = row[50], e51 = row[51];
    for (int cc = 0; cc < WWD; ++cc) {
      const float v = row[cc];
      s1 += v;
      s2 += v * v;
    }
    rw1[0][t] = s1 - e50 - e51;            // cols 0..49
    rw1[1][t] = s1 - e0  - e51;            // cols 1..50
    rw1[2][t] = s1 - e0  - e1;             // cols 2..51
    rw2[0][t] = s2 - e50 * e50 - e51 * e51;
    rw2[1][t] = s2 - e0 * e0   - e51 * e51;
    rw2[2][t] = s2 - e0 * e0   - e1 * e1;
  }
  __syncthreads();

  if (t < KP) {
    const int i = t / 3, j = t % 3;        // patch offset (row i, col j)
    float s1 = 0.f, s2 = 0.f;
    for (int r = i; r < i + OH; ++r) {
      s1 += rw1[j][r];
      s2 += rw2[j][r];
    }
    a1[nc * KP + t] = s1;
    a2[nc * KP + t] = s2;
  }
}

// ---------------------------------------------------------------------------
// Stage 2: single wave32. For each channel c:
//   S1 = A1_c(16xK) * W_c(Kx16),  S2 = A2_c * (W_c^2)   via V_WMMA_F32_16X16X4_F32
//   (K=9 padded to 12 -> 3 WMMA steps per matrix)
// then acc[n,o] += (S2 - S1^2/M) / size[o,c].
// Final: out[o1,n,o2] = acc[n,o2] + B[o1,n].
//
// Fragment layouts (wave32, fp32):
//   A 16x4: lane L(<16): M=L, v0=K0, v1=K1; lane 16+m: M=m, v0=K2, v1=K3
//   B 4x16: v0: lanes0-15=K0, lanes16-31=K2; v1: K1/K3; N = lane&15
//   C/D 16x16: VGPR v: M=v (lanes0-15) / M=v+8 (lanes16-31); N = lane&15
//
// K-padding (9 -> 12) is done branchlessly: clamped index + v_cndmask zero,
// so all global loads stay unconditional (no exec-mask branching).
// ---------------------------------------------------------------------------
__global__ void tvar_stage2(const float* __restrict__ a1g,
                            const float* __restrict__ a2g,
                            const float* __restrict__ Wg,
                            const float* __restrict__ Bg,
                            float* __restrict__ out) {
  const int lane = threadIdx.x;   // 32 threads
  const int m    = lane & 15;     // M row for A, N col for B/C/D
  const int hi   = lane >> 4;
  const float invM = 1.0f / MTOT;

  // Per-lane K indices for the 3 chunks (chunk*4 + hi*2 + {0,1}), padded to 12.
  int   kidx[3][2];
  float kmsk[3][2];
#pragma unroll
  for (int chunk = 0; chunk < 3; ++chunk) {
    const int k0 = chunk * 4 + hi * 2;
    const int k1 = k0 + 1;
    kidx[chunk][0] = (k0 < KP) ? k0 : 0;  kmsk[chunk][0] = (k0 < KP) ? 1.f : 0.f;
    kidx[chunk][1] = (k1 < KP) ? k1 : 0;  kmsk[chunk][1] = (k1 < KP) ? 1.f : 0.f;
  }

  v8f acc = {};

  for (int c = 0; c < CCH; ++c) {
    const int base = (m * CCH + c) * KP;   // m is both n (for A) and o (for B)
    v8f s1 = {};
    v8f s2 = {};
#pragma unroll
    for (int chunk = 0; chunk < 3; ++chunk) {
      const int i0 = base + kidx[chunk][0];
      const int i1 = base + kidx[chunk][1];
      const float m0 = kmsk[chunk][0];
      const float m1 = kmsk[chunk][1];

      const float a10 = a1g[i0] * m0;
      const float a11 = a1g[i1] * m1;
      const float a20 = a2g[i0] * m0;
      const float a21 = a2g[i1] * m1;
      const float w0  = Wg[i0]  * m0;
      const float w1  = Wg[i1]  * m1;

      v2f af1; af1.x = a10; af1.y = a11;
      v2f af2; af2.x = a20; af2.y = a21;
      v2f bf1; bf1.x = w0;       bf1.y = w1;
      v2f bf2; bf2.x = w0 * w0;  bf2.y = w1 * w1;

      s1 = __builtin_amdgcn_wmma_f32_16x16x4_f32(false, af1, false, bf1,
                                                 (short)0, s1, false, false);
      s2 = __builtin_amdgcn_wmma_f32_16x16x4_f32(false, af2, false, bf2,
                                                 (short)0, s2, false, false);
    }

    // per-lane column o = m ; divisor size[o][c] = sum_p W[o][c][p]
    float sz = 0.f;
#pragma unroll
    for (int p = 0; p < KP; ++p) sz += Wg[base + p];
    const float rsz = 1.0f / sz;

#pragma unroll
    for (int v = 0; v < 8; ++v) {
      const float s1v = s1[v];
      const float s2v = s2[v];
      acc[v] += (s2v - s1v * s1v * invM) * rsz;
    }
  }

  // out[o1][n][o2] = acc(M=n, N=o2) + B[o1][n]; lane: o2=m, VGPR v: n=v+8*hi
#pragma unroll
  for (int v = 0; v < 8; ++v) {
    const int n = v + 8 * hi;
    const float val = acc[v];
    for (int o1 = 0; o1 < OO; ++o1) {
      out[(o1 * NN + n) * OO + m] = val + Bg[o1 * CCH + n];
    }
  }
}

extern "C" void kernel_launch(void* const* d_in, const int* in_sizes, int n_in,
                              void* d_out, int out_size, void* d_ws, size_t ws_size,
                              hipStream_t stream) {
  const float* x  = (const float*)d_in[0];   // [16,16,52,52]
  const float* Wg = (const float*)d_in[1];   // [16,16,9,1]
  const float* Bg = (const float*)d_in[2];   // [16,16,1]
  float* out = (float*)d_out;                // [16,16,16]

  float* a1 = (float*)d_ws;                  // [256*9]
  float* a2 = a1 + NN * CCH * KP;            // [256*9]

  tvar_stage1<<<NN * CCH, 64, 0, stream>>>(x, a1, a2);
  tvar_stage2<<<1, 32, 0, stream>>>(a1, a2, Wg, Bg, out);
}